// SparseParticleRouter_7155415515620
// MI455X (gfx1250) — compile-verified
//
#include <hip/hip_runtime.h>

#define B_ 32
#define S_ 16
#define T_ 260
#define D_ 768
#define P_ 256
#define TAIL_ (T_ - P_)   // 4 (bg=1 + extra=3, both copied through unchanged)
#define ROWS_PER_BLK 4    // async-DMA pipeline depth per block

// ---------------------------------------------------------------------------
// Kernel 1: routing. One block per batch element, 256 threads (one/particle).
// Computes scores=max_S z, active mask (+ top-4 fallback), stable compaction
// order via prefix sum, and emits padded_idx / valid / particle_pad_mask.
// ---------------------------------------------------------------------------
__global__ void spr_route_kernel(const float* __restrict__ z,
                                 int* __restrict__ idx_ws,
                                 int* __restrict__ counts_ws,
                                 float* __restrict__ out_mask,
                                 float* __restrict__ out_pidx,
                                 float* __restrict__ out_valid) {
  const int b = blockIdx.x;
  const int t = threadIdx.x;
  __shared__ float s_score[P_];
  __shared__ int   s_flag[P_];
  __shared__ int   s_pre[P_];
  __shared__ int   s_order[P_];
  __shared__ int   s_hasany;

  // score = max over S views
  float sc = -1.0f;
  const float* zb = z + (size_t)b * S_ * P_ + t;
  #pragma unroll
  for (int s = 0; s < S_; ++s) sc = fmaxf(sc, zb[(size_t)s * P_]);
  int fl = (sc > 0.5f) ? 1 : 0;
  s_score[t] = sc;
  s_flag[t]  = fl;
  if (t == 0) s_hasany = 0;
  __syncthreads();
  if (fl) atomicOr(&s_hasany, 1);
  __syncthreads();

  // fallback: top-4 scores (ties -> lower index, matching jax.lax.top_k)
  if (s_hasany == 0) {
    if (t == 0) {
      #pragma unroll
      for (int k = 0; k < 4; ++k) {
        float best = -2.0f; int bi = 0;
        for (int i = 0; i < P_; ++i)
          if (s_score[i] > best) { best = s_score[i]; bi = i; }
        s_flag[bi]  = 1;
        s_score[bi] = -2.0f;
      }
    }
    __syncthreads();
    fl = s_flag[t];
  }

  // inclusive scan of flags (Hillis-Steele over 256 entries)
  s_pre[t] = fl;
  __syncthreads();
  for (int off = 1; off < P_; off <<= 1) {
    int v = (t >= off) ? s_pre[t - off] : 0;
    __syncthreads();
    s_pre[t] += v;
    __syncthreads();
  }
  const int counts = s_pre[P_ - 1];
  const int excl   = s_pre[t] - fl;
  // stable order: active positions ascending, then inactive ascending
  const int rank = fl ? excl : (counts + (t - excl));
  s_order[rank] = t;
  __syncthreads();

  const int pidx = (t < counts) ? s_order[t] : 0;
  const int vld  = (t < counts) ? 1 : 0;

  idx_ws[b * P_ + t]   = pidx;                 // int copy for gather kernel
  out_mask[b * T_ + t] = (float)vld;           // particle_pad_mask[:, :256]
  if (t < TAIL_) out_mask[b * T_ + P_ + t] = 1.0f;  // tail always valid
  out_pidx[b * P_ + t]  = (float)pidx;
  out_valid[b * P_ + t] = (float)vld;
  if (t == 0) counts_ws[b] = counts;
}

// ---------------------------------------------------------------------------
// Kernel 2: active_mean = mean(counts) over B
// ---------------------------------------------------------------------------
__global__ void spr_mean_kernel(const int* __restrict__ counts_ws,
                                float* __restrict__ out_mean) {
  if (threadIdx.x == 0 && blockIdx.x == 0) {
    float s = 0.0f;
    for (int b = 0; b < B_; ++b) s += (float)counts_ws[b];
    *out_mean = s / (float)B_;
  }
}

// ---------------------------------------------------------------------------
// Kernel 3: bandwidth-bound row gather via the CDNA5 async global<->LDS DMA
// path (ASYNCcnt). Each 192-thread block moves ROWS_PER_BLK rows (3 KB each):
// it issues ROWS_PER_BLK b128 async loads per lane into disjoint LDS slots
// (deepening per-wave MLP to 4 outstanding DMAs), one s_wait_asynccnt 0,
// then ROWS_PER_BLK async stores. Row metadata (j, src) is block-uniform,
// so idx_ws lookups scalarize. Lanes own disjoint LDS slots -> no barrier.
// ---------------------------------------------------------------------------
__global__ void __launch_bounds__(192)
spr_gather_kernel(const float* __restrict__ tokens,
                  const int* __restrict__ idx_ws,
                  float* __restrict__ out) {
  __shared__ __align__(16) char buf[ROWS_PER_BLK * 192 * 16];  // 12 KB
  const int t    = threadIdx.x;            // 0..191, 16 B each -> 3 KB/row
  const int base = blockIdx.x * ROWS_PER_BLK;

  char __attribute__((address_space(3)))* lds0 =
      (char __attribute__((address_space(3)))*)buf + t * 16;

  char* gdst[ROWS_PER_BLK];
  #pragma unroll
  for (int r = 0; r < ROWS_PER_BLK; ++r) {
    const int row = base + r;              // b*S*T + s*T + j
    const int j   = row % T_;
    const int bs  = row / T_;              // b*S + s
    const int b   = bs / S_;
    const int src = (j < P_) ? idx_ws[b * P_ + j] : j;   // tail passes through
    const char* gsrc =
        (const char*)(tokens + ((size_t)bs * T_ + src) * D_) + (size_t)t * 16;
    gdst[r] = (char*)(out + ((size_t)bs * T_ + j) * D_) + (size_t)t * 16;
    asm volatile("global_load_async_to_lds_b128 %0, %1, off"
                 :: "v"(lds0 + r * 192 * 16), "v"(gsrc) : "memory");
  }
  asm volatile("s_wait_asynccnt 0x0" ::: "memory");
  #pragma unroll
  for (int r = 0; r < ROWS_PER_BLK; ++r) {
    asm volatile("global_store_async_from_lds_b128 %0, %1, off"
                 :: "v"(gdst[r]), "v"(lds0 + r * 192 * 16) : "memory");
  }
  asm volatile("s_wait_asynccnt 0x0" ::: "memory");
}

// ---------------------------------------------------------------------------
// Launch
// ---------------------------------------------------------------------------
extern "C" void kernel_launch(void* const* d_in, const int* in_sizes, int n_in,
                              void* d_out, int out_size, void* d_ws, size_t ws_size,
                              hipStream_t stream) {
  (void)in_sizes; (void)n_in; (void)out_size; (void)ws_size;
  const float* tokens = (const float*)d_in[0];   // (32,16,260,768) f32
  const float* z      = (const float*)d_in[1];   // (32,16,256) f32
  float* out = (float*)d_out;

  int* idx_ws    = (int*)d_ws;                   // 32*256 ints
  int* counts_ws = idx_ws + B_ * P_;             // 32 ints

  const size_t n_tok = (size_t)B_ * S_ * T_ * D_;
  float* out_mask  = out + n_tok;                       // (32,260)
  float* out_pidx  = out_mask + (size_t)B_ * T_;        // (32,256)
  float* out_valid = out_pidx + (size_t)B_ * P_;        // (32,256)
  float* out_mean  = out_valid + (size_t)B_ * P_;       // scalar

  spr_route_kernel<<<B_, P_, 0, stream>>>(z, idx_ws, counts_ws,
                                          out_mask, out_pidx, out_valid);
  spr_mean_kernel<<<1, 32, 0, stream>>>(counts_ws, out_mean);
  // 133,120 rows / 4 rows per block = 33,280 blocks (divides exactly)
  spr_gather_kernel<<<(B_ * S_ * T_) / ROWS_PER_BLK, 192, 0, stream>>>(
      tokens, idx_ws, out);
}